// YOLOXLoss_79645873537492
// MI455X (gfx1250) — compile-verified
//
#include <hip/hip_runtime.h>
#include <math.h>

// ---------------- problem constants ----------------
#define A_TOT   8400      // 80*80 + 40*40 + 20*20
#define BATCH   32
#define TNUM    50
#define TPAD    64        // targets padded to 4 tiles of 16
#define KSEL    10
#define NCLS    80
#define CHUNK   512       // anchors per phase-1 block (32 WMMA tiles)
#define NCHUNK  17        // ceil(8400/512)
#define NCAND   (NCHUNK*KSEL)   // 170 merge candidates per (b,t)
#define SLOTS   6               // 32 lanes * 6 = 192 >= 170
#define FINF    __builtin_huge_valf()
#define IBIG    0x7FFFFFFF

typedef __attribute__((ext_vector_type(2))) float v2f;
typedef __attribute__((ext_vector_type(8))) float v8f;

// map global anchor id -> per-level index, HW, W, stride
__device__ __forceinline__ void anchor_decode(int a, int& ai, int& HW, int& W, float& s) {
    if (a < 6400)      { ai = a;        HW = 6400; W = 80; s =  8.f; }
    else if (a < 8000) { ai = a - 6400; HW = 1600; W = 40; s = 16.f; }
    else               { ai = a - 8000; HW =  400; W = 20; s = 32.f; }
}

__device__ __forceinline__ float bce_logits(float x, float t) {
    return fmaxf(x, 0.f) - x * t + log1pf(expf(-fabsf(x)));
}

// ---------------- phase 0: clear obj scatter flags ----------------
__global__ void zero_flags(int* __restrict__ objflag, int n) {
    int i = blockIdx.x * blockDim.x + threadIdx.x;
    int stride = gridDim.x * blockDim.x;
    for (; i < n; i += stride) objflag[i] = 0;
}

// ---------------- phase 1: WMMA distance + per-chunk top-10 ----------------
// grid: (NCHUNK, 4 target tiles, BATCH), block: 256 (8 waves).
// Stage A: v_wmma_f32_16x16x4_f32 computes the target·anchor cross terms,
//          epilogue forms dist^2 into LDS [16 targets][512 anchors].
// Stage B: barrier-free per-wave top-10 (2 targets/wave): register-sorted
//          lane-local lists + shfl_xor butterfly merge. Only 2 barriers total.
__global__ void __launch_bounds__(256)
topk_phase1(const float* __restrict__ targets,
            float* __restrict__ topd, int* __restrict__ topi) {
    const int chunk = blockIdx.x;
    const int t16   = blockIdx.y;
    const int b     = blockIdx.z;
    const int tid   = threadIdx.x;
    const int lane  = tid & 31;
    const int wid   = tid >> 5;

    __shared__ float ds_dist[16 * CHUNK];   // 32 KB: dist^2[target][local anchor]
    __shared__ float s_tcx[16], s_tcy[16], s_tn[16];

    if (tid < 16) {
        int tg = t16 * 16 + tid;
        float tcx = 0.f, tcy = 0.f;
        if (tg < TNUM) {
            const float* tp = targets + ((size_t)b * TNUM + tg) * 5;
            tcx = tp[0] + 0.5f * tp[2];
            tcy = tp[1] + 0.5f * tp[3];
        }
        s_tcx[tid] = tcx; s_tcy[tid] = tcy; s_tn[tid] = tcx * tcx + tcy * tcy;
    }
    __syncthreads();

    const int chunkBase = chunk * CHUNK;
    const int n    = lane & 15;     // M (target row) for A, N (anchor col) for B/D
    const int half = lane >> 4;     // K split: lanes 16-31 carry K=2,3 (zeroed)

    // A operand: 16x4 f32; lane m=lane%16 holds (K0,K1) in lanes 0-15, (K2,K3)=0 above
    v2f aop;
    aop.x = (half == 0) ? s_tcx[n] : 0.f;
    aop.y = (half == 0) ? s_tcy[n] : 0.f;

    for (int i = 0; i < 4; ++i) {
        int tt = wid * 4 + i;               // tile 0..31 in this chunk
        int ga = chunkBase + tt * 16 + n;   // global anchor id
        bool valid = ga < A_TOT;
        int aa = valid ? ga : 0;
        int ai, HW, W; float s;
        anchor_decode(aa, ai, HW, W, s);
        float gx = (float)(ai % W), gy = (float)(ai / W);
        float acx = gx * s + 0.5f * s;
        float acy = gy * s + 0.5f * s;
        float an  = acx * acx + acy * acy;

        // B operand: 4x16 f32; lanes 0-15 rows K0,K1; lanes 16-31 rows K2,K3 = 0
        v2f bop;
        bop.x = (half == 0) ? acx : 0.f;
        bop.y = (half == 0) ? acy : 0.f;

        v8f c = {0.f, 0.f, 0.f, 0.f, 0.f, 0.f, 0.f, 0.f};
        v8f d = __builtin_amdgcn_wmma_f32_16x16x4_f32(
            /*neg_a=*/false, aop, /*neg_b=*/false, bop,
            /*c_mod=*/(short)0, c, /*reuse_a=*/false, /*reuse_b=*/false);

        #pragma unroll
        for (int v = 0; v < 8; ++v) {
            int t = v + 8 * half;                 // D layout: M = v + 8*(lane/16)
            float dist2 = s_tn[t] + an - 2.f * d[v];
            ds_dist[t * CHUNK + tt * 16 + n] = valid ? dist2 : FINF;
        }
    }
    __syncthreads();

    // ---- barrier-free top-10: wave `wid` owns targets 2*wid and 2*wid+1 ----
    for (int r = 0; r < 2; ++r) {
        const int t = 2 * wid + r;

        // lane-local sorted top-10 over 16 strided elements (register arrays,
        // fully unrolled compare-swap insertion -> v_cndmask, no scratch)
        float lv[KSEL]; int lj[KSEL];
        #pragma unroll
        for (int q = 0; q < KSEL; ++q) { lv[q] = FINF; lj[q] = IBIG; }

        #pragma unroll
        for (int i = 0; i < CHUNK / 32; ++i) {
            int j = lane + 32 * i;                     // ascending index scan
            float v = ds_dist[t * CHUNK + j];
            bool ins = v < lv[KSEL - 1];               // strict <: stable ties
            lv[KSEL - 1] = ins ? v : lv[KSEL - 1];
            lj[KSEL - 1] = ins ? j : lj[KSEL - 1];
            #pragma unroll
            for (int q = KSEL - 1; q > 0; --q) {       // bubble new entry up
                bool sw = lv[q] < lv[q - 1];
                float tv = lv[q]; int tj = lj[q];
                if (sw) { lv[q] = lv[q-1]; lj[q] = lj[q-1]; lv[q-1] = tv; lj[q-1] = tj; }
            }
        }

        // merge 32 sorted lists -> wave top-10 via butterfly argmin
        const size_t outBase =
            (((size_t)b * TPAD + (t16 * 16 + t)) * NCHUNK + chunk) * KSEL;
        for (int k = 0; k < KSEL; ++k) {
            float v = lv[0]; int j = lj[0]; int owner = lane;
            #pragma unroll
            for (int off = 16; off > 0; off >>= 1) {
                float ov = __shfl_xor(v, off);
                int   oj = __shfl_xor(j, off);
                int   oo = __shfl_xor(owner, off);
                if (ov < v || (ov == v && oj < j)) { v = ov; j = oj; owner = oo; }
            }
            if (lane == 0) {
                topd[outBase + k] = v;
                topi[outBase + k] = (j < CHUNK) ? (chunkBase + j) : 0;
            }
            if (lane == owner) {                       // consume own head
                #pragma unroll
                for (int q = 0; q < KSEL - 1; ++q) { lv[q] = lv[q+1]; lj[q] = lj[q+1]; }
                lv[KSEL - 1] = FINF; lj[KSEL - 1] = IBIG;
            }
        }
    }
}

// ---------------- phase 2: merge + CIoU/cls/obj per (b,t), 1 wave each ----------------
__global__ void __launch_bounds__(32)
assign_phase2(const float* __restrict__ targets,
              const float* __restrict__ p3_cls, const float* __restrict__ p4_cls,
              const float* __restrict__ p5_cls,
              const float* __restrict__ p3_reg, const float* __restrict__ p4_reg,
              const float* __restrict__ p5_reg,
              const float* __restrict__ topd, const int* __restrict__ topi,
              int* __restrict__ objflag,
              float* __restrict__ regp, float* __restrict__ clsp,
              float* __restrict__ wsump) {
    const int bt = blockIdx.x;
    const int b = bt / TNUM, t = bt % TNUM;
    const int lane = threadIdx.x;

    // ---- merge NCAND chunk candidates -> global top-10 (wave butterfly argmin)
    float cd[SLOTS]; int ca[SLOTS];
    const float* tdp = topd + ((size_t)b * TPAD + t) * NCAND;
    const int*   tip = topi + ((size_t)b * TPAD + t) * NCAND;
    #pragma unroll
    for (int s = 0; s < SLOTS; ++s) {
        int j = lane + 32 * s;
        if (j < NCAND) { cd[s] = tdp[j]; ca[s] = tip[j]; }
        else           { cd[s] = FINF;   ca[s] = IBIG; }
    }
    int sel[KSEL];
    for (int k = 0; k < KSEL; ++k) {
        float v = FINF; int aidx = IBIG; int slot = -1;
        #pragma unroll
        for (int s = 0; s < SLOTS; ++s) {
            if (cd[s] < v || (cd[s] == v && ca[s] < aidx)) { v = cd[s]; aidx = ca[s]; slot = s; }
        }
        int owner = lane;
        #pragma unroll
        for (int off = 16; off > 0; off >>= 1) {
            float ov = __shfl_xor(v, off);
            int   oa = __shfl_xor(aidx, off);
            int   oo = __shfl_xor(owner, off);
            int   os = __shfl_xor(slot, off);
            if (ov < v || (ov == v && oa < aidx)) { v = ov; aidx = oa; owner = oo; slot = os; }
        }
        sel[k] = aidx;
        if (lane == owner && slot >= 0) { cd[slot] = FINF; ca[slot] = IBIG; }
    }

    // ---- target box (cx,cy,w,h -> corners)
    const float* tp = targets + ((size_t)b * TNUM + t) * 5;
    float tw = tp[2], th = tp[3];
    float t_cx = tp[0] + 0.5f * tw, t_cy = tp[1] + 0.5f * th;
    int tcls = (int)tp[4];
    float x21 = t_cx - 0.5f * tw, y21 = t_cy - 0.5f * th;
    float x22 = t_cx + 0.5f * tw, y22 = t_cy + 0.5f * th;

    // ---- decode pred boxes + IoU/CIoU (redundant across lanes; loads broadcast)
    float iouv[KSEL], ciouv[KSEL];
    #pragma unroll
    for (int k = 0; k < KSEL; ++k) {
        int a = sel[k];
        int ai, HW, W; float s;
        anchor_decode(a, ai, HW, W, s);
        const float* rp = (a < 6400) ? p3_reg : (a < 8000 ? p4_reg : p5_reg);
        float r0 = rp[((size_t)b * 4 + 0) * HW + ai];
        float r1 = rp[((size_t)b * 4 + 1) * HW + ai];
        float r2 = rp[((size_t)b * 4 + 2) * HW + ai];
        float r3 = rp[((size_t)b * 4 + 3) * HW + ai];
        float gx = (float)(ai % W), gy = (float)(ai / W);
        float pcx = (gx + 1.f / (1.f + expf(-r0))) * s;
        float pcy = (gy + 1.f / (1.f + expf(-r1))) * s;
        float pw = expf(r2) * s, ph = expf(r3) * s;
        float x11 = pcx - 0.5f * pw, y11 = pcy - 0.5f * ph;
        float x12 = pcx + 0.5f * pw, y12 = pcy + 0.5f * ph;
        float iw = fmaxf(fminf(x12, x22) - fmaxf(x11, x21), 0.f);
        float ih = fmaxf(fminf(y12, y22) - fmaxf(y11, y21), 0.f);
        float inter = iw * ih;
        float a1 = (x12 - x11) * (y12 - y11);
        float a2 = (x22 - x21) * (y22 - y21);
        float uni = a1 + a2 - inter + 1e-7f;
        float io = inter / uni;
        iouv[k] = io;
        float cw = fmaxf(x12, x22) - fminf(x11, x21);
        float ch = fmaxf(y12, y22) - fminf(y11, y21);
        float c2 = cw * cw + ch * ch + 1e-7f;
        float ddx = x21 + x22 - x11 - x12, ddy = y21 + y22 - y11 - y12;
        float rho2 = (ddx * ddx + ddy * ddy) * 0.25f;
        float dv = atanf((x22 - x21) / (y22 - y21 + 1e-7f))
                 - atanf((x12 - x11) / (y12 - y11 + 1e-7f));
        float vv = (4.f / (3.14159f * 3.14159f)) * dv * dv;
        float alpha = vv / (1.f - io + vv + 1e-7f);
        ciouv[k] = io - (rho2 / c2 + vv * alpha);
    }

    // ---- assignment mask (valid IoU>thr, else fallback = argmax IoU, first-tie)
    bool anyv = false; float best = -FINF; int am = 0;
    #pragma unroll
    for (int k = 0; k < KSEL; ++k) {
        if (iouv[k] > best) { best = iouv[k]; am = k; }
        anyv = anyv || (iouv[k] > 0.1f);
    }
    float reg_part = 0.f, wsum = 0.f;
    float wk[KSEL];
    #pragma unroll
    for (int k = 0; k < KSEL; ++k) {
        float w = anyv ? ((iouv[k] > 0.1f) ? 1.f : 0.f) : ((k == am) ? 1.f : 0.f);
        wk[k] = w;
        reg_part += (1.f - ciouv[k]) * w;
        wsum += w;
    }

    // ---- classification BCE (lanes parallel over classes) + obj scatter
    float cls_acc = 0.f;
    for (int k = 0; k < KSEL; ++k) {
        if (wk[k] == 0.f) continue;       // uniform across the wave
        int a = sel[k];
        int ai, HW, W; float s;
        anchor_decode(a, ai, HW, W, s);
        const float* cp = (a < 6400) ? p3_cls : (a < 8000 ? p4_cls : p5_cls);
        for (int c = lane; c < NCLS; c += 32) {
            float x = cp[((size_t)b * NCLS + c) * HW + ai];
            cls_acc += bce_logits(x, (c == tcls) ? 1.f : 0.f);
        }
        if (lane == 0) objflag[b * A_TOT + a] = 1;   // all writers write 1: race-free
    }
    #pragma unroll
    for (int off = 16; off > 0; off >>= 1) cls_acc += __shfl_xor(cls_acc, off);

    if (lane == 0) {
        regp[bt] = reg_part;
        clsp[bt] = cls_acc;
        wsump[bt] = wsum;
    }
}

// ---------------- phase 3: objectness BCE over all anchors ----------------
__global__ void __launch_bounds__(256)
obj_phase3(const float* __restrict__ p3_obj, const float* __restrict__ p4_obj,
           const float* __restrict__ p5_obj,
           const int* __restrict__ objflag, float* __restrict__ objp) {
    __shared__ float red[256];
    const int tid = threadIdx.x;
    int gid = blockIdx.x * 256 + tid;
    int stride = gridDim.x * 256;
    float acc = 0.f;
    for (int i = gid; i < BATCH * A_TOT; i += stride) {
        int b = i / A_TOT, a = i % A_TOT;
        int ai, HW, W; float s;
        anchor_decode(a, ai, HW, W, s);
        const float* op = (a < 6400) ? p3_obj : (a < 8000 ? p4_obj : p5_obj);
        float x = op[(size_t)b * HW + ai];
        acc += bce_logits(x, objflag[i] ? 1.f : 0.f);
    }
    red[tid] = acc; __syncthreads();
    for (int sft = 128; sft > 0; sft >>= 1) {
        if (tid < sft) red[tid] += red[tid + sft];
        __syncthreads();
    }
    if (tid == 0) objp[blockIdx.x] = red[0];
}

// ---------------- phase 4: final deterministic reduction ----------------
__global__ void __launch_bounds__(256)
final_reduce(const float* __restrict__ regp, const float* __restrict__ clsp,
             const float* __restrict__ wsump, const float* __restrict__ objp,
             float* __restrict__ out) {
    __shared__ float r0[256], r1[256], r2[256], r3[256];
    const int tid = threadIdx.x;
    float ar = 0.f, ac = 0.f, aw = 0.f;
    for (int i = tid; i < BATCH * TNUM; i += 256) {
        ar += regp[i]; ac += clsp[i]; aw += wsump[i];
    }
    r0[tid] = ar; r1[tid] = ac; r2[tid] = aw; r3[tid] = objp[tid];
    __syncthreads();
    for (int sft = 128; sft > 0; sft >>= 1) {
        if (tid < sft) {
            r0[tid] += r0[tid + sft]; r1[tid] += r1[tid + sft];
            r2[tid] += r2[tid + sft]; r3[tid] += r3[tid + sft];
        }
        __syncthreads();
    }
    if (tid == 0) {
        float num_pos = fmaxf(r2[0], 1.f);
        float reg_l = r0[0] / num_pos;
        float obj_l = r3[0] / num_pos;
        float cls_l = r1[0] / num_pos;
        out[0] = 5.f * reg_l + 1.f * obj_l + 1.f * cls_l;
        out[1] = reg_l;
        out[2] = obj_l;
        out[3] = cls_l;
    }
}

// ---------------- host launcher ----------------
extern "C" void kernel_launch(void* const* d_in, const int* in_sizes, int n_in,
                              void* d_out, int out_size, void* d_ws, size_t ws_size,
                              hipStream_t stream) {
    const float* p3_cls = (const float*)d_in[0];
    const float* p3_reg = (const float*)d_in[1];
    const float* p3_obj = (const float*)d_in[2];
    const float* p4_cls = (const float*)d_in[3];
    const float* p4_reg = (const float*)d_in[4];
    const float* p4_obj = (const float*)d_in[5];
    const float* p5_cls = (const float*)d_in[6];
    const float* p5_reg = (const float*)d_in[7];
    const float* p5_obj = (const float*)d_in[8];
    const float* targets = (const float*)d_in[9];

    // workspace layout (~3.9 MB)
    char* ws = (char*)d_ws;
    size_t off = 0;
    float* topd = (float*)(ws + off); off += (size_t)BATCH * TPAD * NCHUNK * KSEL * 4;
    int*   topi = (int*)(ws + off);   off += (size_t)BATCH * TPAD * NCHUNK * KSEL * 4;
    int*   objflag = (int*)(ws + off); off += (size_t)BATCH * A_TOT * 4;
    float* regp  = (float*)(ws + off); off += (size_t)BATCH * TNUM * 4;
    float* clsp  = (float*)(ws + off); off += (size_t)BATCH * TNUM * 4;
    float* wsump = (float*)(ws + off); off += (size_t)BATCH * TNUM * 4;
    float* objp  = (float*)(ws + off); off += 256 * 4;

    zero_flags<<<256, 256, 0, stream>>>(objflag, BATCH * A_TOT);

    dim3 g1(NCHUNK, 4, BATCH);
    topk_phase1<<<g1, 256, 0, stream>>>(targets, topd, topi);

    assign_phase2<<<BATCH * TNUM, 32, 0, stream>>>(
        targets, p3_cls, p4_cls, p5_cls, p3_reg, p4_reg, p5_reg,
        topd, topi, objflag, regp, clsp, wsump);

    obj_phase3<<<256, 256, 0, stream>>>(p3_obj, p4_obj, p5_obj, objflag, objp);

    final_reduce<<<1, 256, 0, stream>>>(regp, clsp, wsump, objp, (float*)d_out);
}